// CausalBindingMechanism_3453153706265
// MI455X (gfx1250) — compile-verified
//
#include <hip/hip_runtime.h>

typedef float v2f __attribute__((ext_vector_type(2)));
typedef float v8f __attribute__((ext_vector_type(8)));

#define CBM_D    2048
#define CBM_NH   16
#define CBM_HD   128
#define CBM_B    16
#define CBM_NS   8
#define CBM_SEQ  2048
#define INV_SQRT_HD 0.08838834764831845f   // 1/sqrt(128)

// =====================================================================
// RoPE of the broadcast base positional query: pos[s,n], s<SEQ, n<D
// =====================================================================
__global__ void pos_rope_kernel(const float* __restrict__ base, float* __restrict__ pos) {
  int idx = blockIdx.x * blockDim.x + threadIdx.x;       // SEQ*D threads
  int s = idx >> 11;                                     // / D
  int n = idx & (CBM_D - 1);
  int j = n & (CBM_D / 2 - 1);
  // 10000^(-j/1024) = exp(-j * ln(10000)/1024)
  float freq = __expf(-(float)j * (9.210340371976184f / 1024.0f));
  float ang = (float)s * freq;
  float sn, cs;
  __sincosf(ang, &sn, &cs);
  float q1 = base[j];
  float q2 = base[j + CBM_D / 2];
  pos[idx] = (n < CBM_D / 2) ? (q1 * cs - q2 * sn) : (q1 * sn + q2 * cs);
}

// =====================================================================
// C[M,N] = A[M,K] @ B[K,N] + bias[N], f32 WMMA 16x16x4,
// software-pipelined (double-buffered k-steps).
// N, K are COMPILE-TIME so all B accesses are base + immediate offsets.
// grid = (M/16, N/256), block = 256 (8 waves), wave covers 32 cols.
// A-frag (16x4 f32): lanes 0-15 = rows, hold K=k0,k0+1; lanes 16-31 same
// rows hold K=k0+2,k0+3. B-frag mirrors the K split across lane halves.
// Requires K % 8 == 0.
// =====================================================================
template <int N, int K>
__global__ __launch_bounds__(256) void gemm_bias_wmma(const float* __restrict__ A,
                                                      const float* __restrict__ B,
                                                      const float* __restrict__ bias,
                                                      float* __restrict__ C) {
  const int row0  = blockIdx.x * 16;
  const int wave  = threadIdx.x >> 5;
  const int lane  = threadIdx.x & 31;
  const int c0    = blockIdx.y * 256 + wave * 32;
  const int nL    = lane & 15;
  const int kHalf = (lane >> 4) * 2;                     // 0 or 2

  const float* ap = A + (size_t)(row0 + nL) * K + kHalf; // advance 8 floats/iter
  const float* bp = B + (size_t)kHalf * N + c0 + nL;     // advance 8 rows/iter

  v8f acc[2] = {};
  v2f a0, a1, b0[2], b1[2];

  // prologue: k-step 0 resident
  a0 = *(const v2f*)ap;
  b0[0].x = bp[0];  b0[0].y = bp[N];
  b0[1].x = bp[16]; b0[1].y = bp[N + 16];

  for (int it = 0; it < (K >> 3) - 1; ++it) {   // consume 2 k-steps per iteration
    // preload step s+1
    a1 = *(const v2f*)(ap + 4);
    b1[0].x = bp[4 * N];      b1[0].y = bp[5 * N];
    b1[1].x = bp[4 * N + 16]; b1[1].y = bp[5 * N + 16];
    // consume step s
    acc[0] = __builtin_amdgcn_wmma_f32_16x16x4_f32(false, a0, false, b0[0], (short)0, acc[0], false, false);
    acc[1] = __builtin_amdgcn_wmma_f32_16x16x4_f32(false, a0, false, b0[1], (short)0, acc[1], false, false);
    ap += 8;
    bp += 8 * (size_t)N;
    // preload step s+2
    a0 = *(const v2f*)ap;
    b0[0].x = bp[0];  b0[0].y = bp[N];
    b0[1].x = bp[16]; b0[1].y = bp[N + 16];
    // consume step s+1
    acc[0] = __builtin_amdgcn_wmma_f32_16x16x4_f32(false, a1, false, b1[0], (short)0, acc[0], false, false);
    acc[1] = __builtin_amdgcn_wmma_f32_16x16x4_f32(false, a1, false, b1[1], (short)0, acc[1], false, false);
  }
  // epilogue: steps K/4-2 (resident) and K/4-1
  a1 = *(const v2f*)(ap + 4);
  b1[0].x = bp[4 * N];      b1[0].y = bp[5 * N];
  b1[1].x = bp[4 * N + 16]; b1[1].y = bp[5 * N + 16];
  acc[0] = __builtin_amdgcn_wmma_f32_16x16x4_f32(false, a0, false, b0[0], (short)0, acc[0], false, false);
  acc[1] = __builtin_amdgcn_wmma_f32_16x16x4_f32(false, a0, false, b0[1], (short)0, acc[1], false, false);
  acc[0] = __builtin_amdgcn_wmma_f32_16x16x4_f32(false, a1, false, b1[0], (short)0, acc[0], false, false);
  acc[1] = __builtin_amdgcn_wmma_f32_16x16x4_f32(false, a1, false, b1[1], (short)0, acc[1], false, false);

  // C/D layout: VGPR v, lanes 0-15 -> row v, lanes 16-31 -> row v+8; col = lane&15
  const int rowOff = (lane >> 4) * 8;
#pragma unroll
  for (int t = 0; t < 2; ++t) {
    const int n = c0 + t * 16 + nL;
    const float bn = bias[n];
#pragma unroll
    for (int v = 0; v < 8; ++v)
      C[(size_t)(row0 + rowOff + v) * N + n] = acc[t][v] + bn;
  }
}

// =====================================================================
// Deterministic block LayerNorm stats for 256-thread blocks (8 waves)
// =====================================================================
__device__ __forceinline__ void block_ln_stats256(float ps, float pq, float* red,
                                                  float* mean_out, float* rstd_out) {
#pragma unroll
  for (int m = 16; m >= 1; m >>= 1) {
    ps += __shfl_xor(ps, m, 32);
    pq += __shfl_xor(pq, m, 32);
  }
  const int wave = threadIdx.x >> 5;
  const int lane = threadIdx.x & 31;
  if (lane == 0) { red[wave] = ps; red[8 + wave] = pq; }
  __syncthreads();
  float s = 0.f, q = 0.f;
#pragma unroll
  for (int w = 0; w < 8; ++w) { s += red[w]; q += red[8 + w]; }
  float mean = s * (1.0f / CBM_D);
  float var  = q * (1.0f / CBM_D) - mean * mean;
  *mean_out = mean;
  *rstd_out = rsqrtf(var + 1e-5f);
}

// bound1 = LN(slots + bind_out_broadcast); one block per row (B*NS rows)
__global__ void bind_add_ln_kernel(const float* __restrict__ slots,
                                   const float* __restrict__ bindo,
                                   const float* __restrict__ g,
                                   const float* __restrict__ bvec,
                                   float* __restrict__ out) {
  __shared__ float xs[CBM_D];
  __shared__ float red[16];
  const int row = blockIdx.x;
  const int b = row >> 3;
  float ps = 0.f, pq = 0.f;
  for (int d = threadIdx.x; d < CBM_D; d += 256) {
    float x = slots[(size_t)row * CBM_D + d] + bindo[(size_t)b * CBM_D + d];
    xs[d] = x; ps += x; pq += x * x;
  }
  float mean, rstd;
  block_ln_stats256(ps, pq, red, &mean, &rstd);   // contains the needed barrier
  for (int d = threadIdx.x; d < CBM_D; d += 256)
    out[(size_t)row * CBM_D + d] = (xs[d] - mean) * rstd * g[d] + bvec[d];
}

// interv self-attention over 8 slots; one thread per (b, slot, head)
__global__ void interv_attn_kernel(const float* __restrict__ Q,
                                   const float* __restrict__ K,
                                   const float* __restrict__ V,
                                   float* __restrict__ msgh) {
  const int t   = blockIdx.x * blockDim.x + threadIdx.x;  // B*NS*NH
  const int h   = t & (CBM_NH - 1);
  const int row = t >> 4;                                 // b*NS + i
  const int b   = row >> 3;
  const float* q  = Q + (size_t)row * CBM_D + h * CBM_HD;
  const float* Kb = K + ((size_t)b * CBM_NS) * CBM_D + h * CBM_HD;
  const float* Vb = V + ((size_t)b * CBM_NS) * CBM_D + h * CBM_HD;
  float sc[CBM_NS] = {};
  for (int u = 0; u < CBM_HD; ++u) {
    float qv = q[u];
#pragma unroll
    for (int j = 0; j < CBM_NS; ++j) sc[j] += qv * Kb[(size_t)j * CBM_D + u];
  }
  float mx = -1e30f;
#pragma unroll
  for (int j = 0; j < CBM_NS; ++j) { sc[j] *= INV_SQRT_HD; mx = fmaxf(mx, sc[j]); }
  float den = 0.f;
#pragma unroll
  for (int j = 0; j < CBM_NS; ++j) { sc[j] = __expf(sc[j] - mx); den += sc[j]; }
  float inv = 1.0f / den;
#pragma unroll
  for (int j = 0; j < CBM_NS; ++j) sc[j] *= inv;
  float* dst = msgh + (size_t)row * CBM_D + h * CBM_HD;
  for (int u = 0; u < CBM_HD; ++u) {
    float a = 0.f;
#pragma unroll
    for (int j = 0; j < CBM_NS; ++j) a += sc[j] * Vb[(size_t)j * CBM_D + u];
    dst[u] = a;
  }
}

// bound2 = LN(bound1 + edge_weights @ messages); one block per row
__global__ void interv_combine_ln_kernel(const float* __restrict__ bound1,
                                         const float* __restrict__ msgs,
                                         const float* __restrict__ ew,
                                         const float* __restrict__ g,
                                         const float* __restrict__ bvec,
                                         float* __restrict__ out) {
  __shared__ float xs[CBM_D];
  __shared__ float red[16];
  const int row = blockIdx.x;
  const int b = row >> 3, i = row & 7;
  float w[CBM_NS];
#pragma unroll
  for (int j = 0; j < CBM_NS; ++j) w[j] = ew[(size_t)b * 64 + i * 8 + j];
  float ps = 0.f, pq = 0.f;
  for (int d = threadIdx.x; d < CBM_D; d += 256) {
    float x = bound1[(size_t)row * CBM_D + d];
#pragma unroll
    for (int j = 0; j < CBM_NS; ++j) x += w[j] * msgs[((size_t)b * CBM_NS + j) * CBM_D + d];
    xs[d] = x; ps += x; pq += x * x;
  }
  float mean, rstd;
  block_ln_stats256(ps, pq, red, &mean, &rstd);
  for (int d = threadIdx.x; d < CBM_D; d += 256)
    out[(size_t)row * CBM_D + d] = (xs[d] - mean) * rstd * g[d] + bvec[d];
}

// =====================================================================
// Fused bcast: 8-key attention -> ctx tile (LDS) -> WMMA GEMM vs Wo
// -> +bias +pos -> LayerNorm -> out.
// 16 rows/WG, 256 threads = 8 waves (2 waves/SIMD -> ~256 VGPR budget);
// each wave owns 16 rows x 256 cols = 16 accumulator tiles (128 VGPRs)
// + 64 VGPRs double-buffered B fragments: fits without scratch spills.
// LDS row stride padded to 2052 words so the 16 per-lane A-fragment rows
// map to distinct banks (2052 % 64 == 4).
// =====================================================================
#define LDSTRIDE 2052
#define FUSED_LDS_BYTES ((16 * LDSTRIDE + 32) * sizeof(float))

__global__ __launch_bounds__(256, 2)
void bcast_fused_kernel(const float* __restrict__ Qpos,  // [SEQ,D]
                        const float* __restrict__ Kc,    // [B,NS,D]
                        const float* __restrict__ Vc,    // [B,NS,D]
                        const float* __restrict__ Wo,    // [D,D]
                        const float* __restrict__ bo,    // [D]
                        const float* __restrict__ pos,   // [SEQ,D]
                        const float* __restrict__ g,
                        const float* __restrict__ bvec,
                        float* __restrict__ out) {       // [B,SEQ,D]
  extern __shared__ float lds[];
  float* ctx   = lds;                   // [16][LDSTRIDE]
  float* statM = lds + 16 * LDSTRIDE;   // [16]
  float* statR = statM + 16;            // [16]

  const int tid  = threadIdx.x;
  const int wave = tid >> 5;
  const int lane = tid & 31;
  const int row0 = blockIdx.x * 16;          // 2048 % 16 == 0 -> single batch b
  const int b    = row0 / CBM_SEQ;

  // ---- Phase 1: attention over 8 keys, one thread per (row, head) ----
  {
    const int r = tid & 15;
    const int h = tid >> 4;
    const int s = (row0 + r) & (CBM_SEQ - 1);
    const float* q  = Qpos + (size_t)s * CBM_D + h * CBM_HD;
    const float* Kb = Kc + ((size_t)b * CBM_NS) * CBM_D + h * CBM_HD;
    const float* Vb = Vc + ((size_t)b * CBM_NS) * CBM_D + h * CBM_HD;
    float sc[CBM_NS] = {};
    for (int u = 0; u < CBM_HD; ++u) {
      float qv = q[u];
#pragma unroll
      for (int j = 0; j < CBM_NS; ++j) sc[j] += qv * Kb[(size_t)j * CBM_D + u];
    }
    float mx = -1e30f;
#pragma unroll
    for (int j = 0; j < CBM_NS; ++j) { sc[j] *= INV_SQRT_HD; mx = fmaxf(mx, sc[j]); }
    float den = 0.f;
#pragma unroll
    for (int j = 0; j < CBM_NS; ++j) { sc[j] = __expf(sc[j] - mx); den += sc[j]; }
    float inv = 1.0f / den;
#pragma unroll
    for (int j = 0; j < CBM_NS; ++j) sc[j] *= inv;
    float* dst = ctx + (size_t)r * LDSTRIDE + h * CBM_HD;
    for (int u = 0; u < CBM_HD; ++u) {
      float a = 0.f;
#pragma unroll
      for (int j = 0; j < CBM_NS; ++j) a += sc[j] * Vb[(size_t)j * CBM_D + u];
      dst[u] = a;
    }
  }
  __syncthreads();

  // ---- Phase 2: ctx(16,2048) @ Wo with f32 WMMA, double-buffered.
  // Wave w: rows 0..15, cols c0..c0+255 -> 16 accumulator tiles.
  const int c0    = wave * 256;
  const int nL    = lane & 15;
  const int kHalf = (lane >> 4) * 2;

  const float* ap = ctx + (size_t)nL * LDSTRIDE + kHalf;   // LDS, +8/iter
  const float* bp = Wo + (size_t)kHalf * CBM_D + c0 + nL;  // global, +8 rows/iter

  v8f acc[16] = {};
  v2f a0, a1, b0[16], b1[16];

  a0 = *(const v2f*)ap;
#pragma unroll
  for (int t = 0; t < 16; ++t) { b0[t].x = bp[t * 16]; b0[t].y = bp[CBM_D + t * 16]; }

  for (int it = 0; it < (CBM_D >> 3) - 1; ++it) {   // 255 iterations, 2 k-steps each
    // preload step s+1
    a1 = *(const v2f*)(ap + 4);
#pragma unroll
    for (int t = 0; t < 16; ++t) { b1[t].x = bp[4 * CBM_D + t * 16]; b1[t].y = bp[5 * CBM_D + t * 16]; }
    // consume step s
#pragma unroll
    for (int t = 0; t < 16; ++t)
      acc[t] = __builtin_amdgcn_wmma_f32_16x16x4_f32(false, a0, false, b0[t], (short)0, acc[t], false, false);
    ap += 8;
    bp += 8 * CBM_D;
    // preload step s+2
    a0 = *(const v2f*)ap;
#pragma unroll
    for (int t = 0; t < 16; ++t) { b0[t].x = bp[t * 16]; b0[t].y = bp[CBM_D + t * 16]; }
    // consume step s+1
#pragma unroll
    for (int t = 0; t < 16; ++t)
      acc[t] = __builtin_amdgcn_wmma_f32_16x16x4_f32(false, a1, false, b1[t], (short)0, acc[t], false, false);
  }
  // epilogue: final two k-steps
  a1 = *(const v2f*)(ap + 4);
#pragma unroll
  for (int t = 0; t < 16; ++t) { b1[t].x = bp[4 * CBM_D + t * 16]; b1[t].y = bp[5 * CBM_D + t * 16]; }
#pragma unroll
  for (int t = 0; t < 16; ++t)
    acc[t] = __builtin_amdgcn_wmma_f32_16x16x4_f32(false, a0, false, b0[t], (short)0, acc[t], false, false);
#pragma unroll
  for (int t = 0; t < 16; ++t)
    acc[t] = __builtin_amdgcn_wmma_f32_16x16x4_f32(false, a1, false, b1[t], (short)0, acc[t], false, false);

  __syncthreads();        // all ctx reads finished; reuse LDS for residual

  // ---- Phase 3a: x = bc + bias + pos  -> LDS ----
  {
    const int rowBase = (lane >> 4) * 8;
#pragma unroll
    for (int t = 0; t < 16; ++t) {
      const int n = c0 + t * 16 + nL;
      const float bn = bo[n];
#pragma unroll
      for (int v = 0; v < 8; ++v) {
        const int r = rowBase + v;
        const int s = (row0 + r) & (CBM_SEQ - 1);
        ctx[(size_t)r * LDSTRIDE + n] = acc[t][v] + bn + pos[(size_t)s * CBM_D + n];
      }
    }
  }
  __syncthreads();

  // ---- Phase 3b: deterministic per-row LN stats (wave w -> rows 2w,2w+1) ----
#pragma unroll
  for (int rr = 0; rr < 2; ++rr) {
    const int r = wave * 2 + rr;
    float s = 0.f, qs = 0.f;
    for (int n = lane; n < CBM_D; n += 32) {
      float x = ctx[(size_t)r * LDSTRIDE + n];
      s += x; qs += x * x;
    }
#pragma unroll
    for (int m = 16; m >= 1; m >>= 1) {
      s  += __shfl_xor(s,  m, 32);
      qs += __shfl_xor(qs, m, 32);
    }
    if (lane == 0) {
      float mean = s * (1.0f / CBM_D);
      float var  = qs * (1.0f / CBM_D) - mean * mean;
      statM[r] = mean;
      statR[r] = rsqrtf(var + 1e-5f);
    }
  }
  __syncthreads();

  // ---- Phase 3c: normalize + affine, coalesced store ----
  for (int idx = tid; idx < 16 * CBM_D; idx += 256) {
    const int r = idx >> 11;
    const int n = idx & (CBM_D - 1);
    float x = ctx[(size_t)r * LDSTRIDE + n];
    out[(size_t)(row0 + r) * CBM_D + n] = (x - statM[r]) * statR[r] * g[n] + bvec[n];
  }
}

// =====================================================================
// Host orchestration
// =====================================================================
extern "C" void kernel_launch(void* const* d_in, const int* in_sizes, int n_in,
                              void* d_out, int out_size, void* d_ws, size_t ws_size,
                              hipStream_t stream) {
  (void)in_sizes; (void)n_in; (void)out_size; (void)ws_size;
  auto in = [&](int i) { return (const float*)d_in[i]; };

  const float* slots   = in(0);   // (B,NS,D)
  const float* content = in(1);   // (B,D)
  const float* ew      = in(2);   // (B,NS,NS)
  // binding q/k (idx 3..6) are mathematically dead: softmax over 1 key == 1
  const float* Wv_bind = in(7),  *bv_bind = in(8);
  const float* Wo_bind = in(9),  *bo_bind = in(10);
  const float* Wq_int  = in(11), *bq_int  = in(12);
  const float* Wk_int  = in(13), *bk_int  = in(14);
  const float* Wv_int  = in(15), *bv_int  = in(16);
  const float* Wo_int  = in(17), *bo_int  = in(18);
  const float* Wq_bc   = in(19), *bq_bc   = in(20);
  const float* Wk_bc   = in(21), *bk_bc   = in(22);
  const float* Wv_bc   = in(23), *bv_bc   = in(24);
  const float* Wo_bc   = in(25), *bo_bc   = in(26);
  const float* g_bind  = in(27), *b_bind  = in(28);
  const float* g_int   = in(29), *b_int   = in(30);
  const float* g_bc    = in(31), *b_bc    = in(32);
  const float* baseq   = in(33);  // (D,)
  // in(34) = seq_len scalar on device; SEQ is a module constant (2048)

  float* ws = (float*)d_ws;
  size_t o = 0;
  float* pos    = ws + o; o += (size_t)CBM_SEQ * CBM_D;        // 16 MB
  float* qpos   = ws + o; o += (size_t)CBM_SEQ * CBM_D;        // 16 MB
  float* bindv  = ws + o; o += (size_t)CBM_B * CBM_D;
  float* bindo  = ws + o; o += (size_t)CBM_B * CBM_D;
  const size_t RS = (size_t)CBM_B * CBM_NS * CBM_D;            // 128 x 2048
  float* bound1 = ws + o; o += RS;
  float* Qi     = ws + o; o += RS;
  float* Ki     = ws + o; o += RS;
  float* Vi     = ws + o; o += RS;
  float* msgh   = ws + o; o += RS;
  float* msgs   = ws + o; o += RS;
  float* bound2 = ws + o; o += RS;
  float* Kc     = ws + o; o += RS;
  float* Vc     = ws + o; o += RS;

  float* outp = (float*)d_out;

  const dim3 blk256(256);
  const dim3 gSmall(1, 8);     // M = 16
  const dim3 gRow(8, 8);       // M = 128
  const dim3 gQpos(128, 8);    // M = 2048

  // pos = rope(broadcast base query)
  pos_rope_kernel<<<(CBM_SEQ * CBM_D) / 256, blk256, 0, stream>>>(baseq, pos);

  // Shared-across-batch query projection (2048^3 GEMM, f32 WMMA)
  gemm_bias_wmma<CBM_D, CBM_D><<<gQpos, blk256, 0, stream>>>(pos, Wq_bc, bq_bc, qpos);

  // binding path (single-key attention collapses to V -> O projections)
  gemm_bias_wmma<CBM_D, CBM_D><<<gSmall, blk256, 0, stream>>>(content, Wv_bind, bv_bind, bindv);
  gemm_bias_wmma<CBM_D, CBM_D><<<gSmall, blk256, 0, stream>>>(bindv, Wo_bind, bo_bind, bindo);
  bind_add_ln_kernel<<<CBM_B * CBM_NS, blk256, 0, stream>>>(slots, bindo, g_bind, b_bind, bound1);

  // interv MHA
  gemm_bias_wmma<CBM_D, CBM_D><<<gRow, blk256, 0, stream>>>(bound1, Wq_int, bq_int, Qi);
  gemm_bias_wmma<CBM_D, CBM_D><<<gRow, blk256, 0, stream>>>(bound1, Wk_int, bk_int, Ki);
  gemm_bias_wmma<CBM_D, CBM_D><<<gRow, blk256, 0, stream>>>(bound1, Wv_int, bv_int, Vi);
  interv_attn_kernel<<<(CBM_B * CBM_NS * CBM_NH) / 256, blk256, 0, stream>>>(Qi, Ki, Vi, msgh);
  gemm_bias_wmma<CBM_D, CBM_D><<<gRow, blk256, 0, stream>>>(msgh, Wo_int, bo_int, msgs);
  interv_combine_ln_kernel<<<CBM_B * CBM_NS, blk256, 0, stream>>>(bound1, msgs, ew, g_int, b_int, bound2);

  // bcast K/V projections
  gemm_bias_wmma<CBM_D, CBM_D><<<gRow, blk256, 0, stream>>>(bound2, Wk_bc, bk_bc, Kc);
  gemm_bias_wmma<CBM_D, CBM_D><<<gRow, blk256, 0, stream>>>(bound2, Wv_bc, bv_bc, Vc);

  // Fused attention + output projection + residual + LayerNorm
  bcast_fused_kernel<<<(CBM_B * CBM_SEQ) / 16, blk256, FUSED_LDS_BYTES, stream>>>(
      qpos, Kc, Vc, Wo_bc, bo_bc, pos, g_bc, b_bc, outp);
}